// QuantumQLSTM_65481071395333
// MI455X (gfx1250) — compile-verified
//
#include <hip/hip_runtime.h>

typedef float v2f __attribute__((ext_vector_type(2)));
typedef float v8f __attribute__((ext_vector_type(8)));

#define T_STEPS 256
#define BATCH   1024
#define DIN     128
#define HID     4
#define NG      16                    // 4 gates * 4 hidden units
#define MROWS   (T_STEPS * BATCH)     // 262144 rows of the big GEMM
#define TILES_PER_WAVE 4
#define WAVES_PER_BLOCK 8

// ---------------------------------------------------------------------------
// Kernel 1: split each gate's W[4,132] into Wx[4,128] / Wh[4,4], pack the
// four gates into Wx[16][128], Wh[16][4], bias[16], th[16].
// ---------------------------------------------------------------------------
__device__ __forceinline__ const float* sel4(int g, const float* a, const float* b,
                                             const float* c, const float* d) {
  return g == 0 ? a : (g == 1 ? b : (g == 2 ? c : d));
}

__global__ void qlstm_pack_kernel(
    const float* __restrict__ Wf, const float* __restrict__ bf, const float* __restrict__ thf,
    const float* __restrict__ Wi, const float* __restrict__ bi, const float* __restrict__ thi,
    const float* __restrict__ Wu, const float* __restrict__ bu, const float* __restrict__ thu,
    const float* __restrict__ Wo, const float* __restrict__ bo, const float* __restrict__ tho,
    float* __restrict__ Wx,    // [16][128]
    float* __restrict__ Wh,    // [16][4]
    float* __restrict__ bias,  // [16]
    float* __restrict__ th)    // [16]
{
  for (int idx = threadIdx.x; idx < NG * DIN; idx += blockDim.x) {
    int n = idx >> 7, k = idx & 127;
    const float* W = sel4(n >> 2, Wf, Wi, Wu, Wo);
    Wx[idx] = W[(n & 3) * (DIN + HID) + k];
  }
  for (int idx = threadIdx.x; idx < NG * HID; idx += blockDim.x) {
    int n = idx >> 2, m = idx & 3;
    const float* W = sel4(n >> 2, Wf, Wi, Wu, Wo);
    Wh[idx] = W[(n & 3) * (DIN + HID) + DIN + m];
  }
  if (threadIdx.x < NG) {
    int n = threadIdx.x;
    const float* B = sel4(n >> 2, bf, bi, bu, bo);
    const float* T = sel4(n >> 2, thf, thi, thu, tho);
    bias[n] = B[n & 3];
    th[n]   = T[n & 3];
  }
}

// ---------------------------------------------------------------------------
// Kernel 2: P[M,16] = X[M,128] @ Wx^T[128,16] + bias, via V_WMMA_F32_16X16X4_F32.
// One wave computes TILES_PER_WAVE consecutive 16x16 output tiles; the 32 B
// fragments (8 KB of weights) are loaded once per wave and stay in VGPRs
// across the tile loop, so steady state is 1 A-load (b64) + 1 WMMA per k-step.
// A fragment (16x4 f32): lane L<16 -> row M=L, K={0,1}; lane L+16 -> K={2,3}.
// B fragment (4x16 f32): lane n<16 -> col N=n, K={0,1}; lane n+16 -> K={2,3};
//   B[k][n] = Wx[n][k], so both fragments are aligned float2 loads.
// ---------------------------------------------------------------------------
__global__ __launch_bounds__(WAVES_PER_BLOCK * 32) void qlstm_pregemm_kernel(
    const float* __restrict__ X,     // [MROWS][128]
    const float* __restrict__ Wx,    // [16][128]
    const float* __restrict__ bias,  // [16]
    float* __restrict__ P)           // [MROWS][16]
{
  const int wave     = threadIdx.x >> 5;
  const int lane     = threadIdx.x & 31;
  const int tileBase = (blockIdx.x * WAVES_PER_BLOCK + wave) * TILES_PER_WAVE;
  const int n        = lane & 15;
  const int koff     = (lane >> 4) << 1;   // 0 for lanes 0-15, 2 for lanes 16-31
  const int mHalf    = (lane >> 4) << 3;   // +8 rows for upper half-wave

  // Preload all 32 B fragments once per wave (loop-invariant across tiles).
  v2f bfrag[32];
#pragma unroll
  for (int kk = 0; kk < 32; ++kk)
    bfrag[kk] = *(const v2f*)(Wx + n * DIN + (kk << 2) + koff);

  const float bv = bias[n];

  for (int i = 0; i < TILES_PER_WAVE; ++i) {
    const int    rowBase = (tileBase + i) * 16;
    const float* arow    = X + (size_t)(rowBase + n) * DIN + koff;

    // Prefetch the next tile's A row into cache while this tile computes.
    if (i + 1 < TILES_PER_WAVE)
      __builtin_prefetch(arow + 16 * DIN, 0, 3);

    v8f acc = {0.f, 0.f, 0.f, 0.f, 0.f, 0.f, 0.f, 0.f};
#pragma unroll
    for (int kk = 0; kk < 32; ++kk) {
      v2f a = *(const v2f*)(arow + (kk << 2));
      acc = __builtin_amdgcn_wmma_f32_16x16x4_f32(
          /*neg_a=*/false, a, /*neg_b=*/false, bfrag[kk],
          /*c_mod=*/(short)0, acc, /*reuse_a=*/false, /*reuse_b=*/false);
    }

    const int mBase = rowBase + mHalf;
#pragma unroll
    for (int r = 0; r < 8; ++r)
      P[(size_t)(mBase + r) * NG + n] = acc[r] + bv;
  }
}

// ---------------------------------------------------------------------------
// Kernel 3: sequential scan over T. One thread per batch element.
// v[n] = P[t,b,n] + Wh[n,:] . h ;  c = cos(v + th) ; qgate products ;
// f,i,o = sigmoid, u = tanh ; LSTM cell update ; emit h per step.
// ---------------------------------------------------------------------------
__device__ __forceinline__ float fsigmoid(float x) { return 1.0f / (1.0f + __expf(-x)); }
__device__ __forceinline__ float ftanh(float x)    { return 1.0f - 2.0f / (1.0f + __expf(2.0f * x)); }

__global__ __launch_bounds__(128) void qlstm_scan_kernel(
    const float* __restrict__ P,    // [T][B][16]
    const float* __restrict__ Wh,   // [16][4]
    const float* __restrict__ th,   // [16]
    float* __restrict__ outSeq,     // [T][B][4]
    float* __restrict__ hOut,       // [B][4]
    float* __restrict__ cOut)       // [B][4]
{
  __shared__ float sWh[NG * HID];
  __shared__ float sTh[NG];
  const int tid = threadIdx.x;
  if (tid < NG * HID) sWh[tid] = Wh[tid];
  if (tid < NG)       sTh[tid] = th[tid];
  __syncthreads();

  const int b = blockIdx.x * blockDim.x + tid;

  float h[4] = {0.f, 0.f, 0.f, 0.f};
  float c[4] = {0.f, 0.f, 0.f, 0.f};

  for (int t = 0; t < T_STEPS; ++t) {
    const float4* pv = (const float4*)(P + ((size_t)t * BATCH + b) * NG);
    float4 g0 = pv[0], g1 = pv[1], g2 = pv[2], g3 = pv[3];
    float v[NG] = {g0.x, g0.y, g0.z, g0.w, g1.x, g1.y, g1.z, g1.w,
                   g2.x, g2.y, g2.z, g2.w, g3.x, g3.y, g3.z, g3.w};
#pragma unroll
    for (int nn = 0; nn < NG; ++nn)
      v[nn] += sWh[nn * 4 + 0] * h[0] + sWh[nn * 4 + 1] * h[1] +
               sWh[nn * 4 + 2] * h[2] + sWh[nn * 4 + 3] * h[3];

    float q[NG];
#pragma unroll
    for (int g = 0; g < 4; ++g) {
      float c0 = __cosf(v[g * 4 + 0] + sTh[g * 4 + 0]);
      float c1 = __cosf(v[g * 4 + 1] + sTh[g * 4 + 1]);
      float c2 = __cosf(v[g * 4 + 2] + sTh[g * 4 + 2]);
      float c3 = __cosf(v[g * 4 + 3] + sTh[g * 4 + 3]);
      q[g * 4 + 0] = c1 * c2 * c3;
      q[g * 4 + 1] = c0 * c1;
      q[g * 4 + 2] = c0 * c1 * c2;
      q[g * 4 + 3] = c0 * c1 * c2 * c3;
    }
#pragma unroll
    for (int j = 0; j < 4; ++j) {
      float fg = fsigmoid(q[0 * 4 + j]);   // forget
      float ig = fsigmoid(q[1 * 4 + j]);   // input
      float ug = ftanh(q[2 * 4 + j]);      // update (tanh)
      float og = fsigmoid(q[3 * 4 + j]);   // output
      c[j] = fg * c[j] + ig * ug;
      h[j] = og * ftanh(c[j]);
    }
    *(float4*)(outSeq + ((size_t)t * BATCH + b) * HID) =
        make_float4(h[0], h[1], h[2], h[3]);
  }
  *(float4*)(hOut + b * HID) = make_float4(h[0], h[1], h[2], h[3]);
  *(float4*)(cOut + b * HID) = make_float4(c[0], c[1], c[2], c[3]);
}

// ---------------------------------------------------------------------------
extern "C" void kernel_launch(void* const* d_in, const int* in_sizes, int n_in,
                              void* d_out, int out_size, void* d_ws, size_t ws_size,
                              hipStream_t stream) {
  const float* X   = (const float*)d_in[0];
  const float* Wf  = (const float*)d_in[1];
  const float* bf  = (const float*)d_in[2];
  const float* thf = (const float*)d_in[3];
  const float* Wi  = (const float*)d_in[4];
  const float* bi  = (const float*)d_in[5];
  const float* thi = (const float*)d_in[6];
  const float* Wu  = (const float*)d_in[7];
  const float* bu  = (const float*)d_in[8];
  const float* thu = (const float*)d_in[9];
  const float* Wo  = (const float*)d_in[10];
  const float* bo  = (const float*)d_in[11];
  const float* tho = (const float*)d_in[12];

  // Workspace layout: P (16 MB) | Wx (8 KB) | Wh (256 B) | bias (64 B) | th (64 B)
  char*  ws   = (char*)d_ws;
  float* P    = (float*)ws;
  float* Wx   = (float*)(ws + (size_t)MROWS * NG * sizeof(float));
  float* Wh   = Wx + NG * DIN;
  float* bias = Wh + NG * HID;
  float* th   = bias + NG;

  qlstm_pack_kernel<<<1, 128, 0, stream>>>(Wf, bf, thf, Wi, bi, thi,
                                           Wu, bu, thu, Wo, bo, tho,
                                           Wx, Wh, bias, th);

  // 16384 tiles / (4 tiles per wave * 8 waves per block) = 512 blocks
  const int nTiles = MROWS / 16;
  qlstm_pregemm_kernel<<<nTiles / (TILES_PER_WAVE * WAVES_PER_BLOCK),
                         WAVES_PER_BLOCK * 32, 0, stream>>>(X, Wx, bias, P);

  float* out = (float*)d_out;
  float* seq = out;                                // [256][1024][4]
  float* hO  = out + (size_t)T_STEPS * BATCH * HID;
  float* cO  = hO + (size_t)BATCH * HID;
  qlstm_scan_kernel<<<BATCH / 128, 128, 0, stream>>>(P, Wh, th, seq, hO, cO);
}